// SRKAN_68332929679789
// MI455X (gfx1250) — compile-verified
//
#include <hip/hip_runtime.h>
#include <hip/hip_bf16.h>
#include <math.h>

// Problem constants
#define BB 256
#define TT 1024
#define DD 128
#define LL 16
#define OO 256

typedef __attribute__((ext_vector_type(2))) float v2f;
typedef __attribute__((ext_vector_type(4))) float v4f;
typedef __attribute__((ext_vector_type(8))) float v8f;

// ---------------------------------------------------------------------------
// Kernel 1: agg[bt, o] = sum_l act_l( x[bt,:] . sub_W[l,:] + sub_b[l] ) * agg_W[l,o]
// One workgroup = 64 (b,t) rows. Stage 1: VALU dots + activations into LDS.
// Stage 2: 16-K WMMA f32 aggregation matmul, result -> d_out (holds agg).
// ---------------------------------------------------------------------------
__global__ __launch_bounds__(256) void srkan_agg_kernel(
    const float* __restrict__ x, const float* __restrict__ subW,
    const float* __restrict__ subB, const float* __restrict__ aggW,
    float* __restrict__ out)
{
    __shared__ float xs[64 * 132];   // 64 rows of x, padded to 132 (bank spread)
    __shared__ float sw[16 * 132];   // sub_W padded
    __shared__ float acts[64 * 20];  // activation tile, row pad 20
    __shared__ float awz[4096];      // agg_W pair-swizzled: [(k>>1)*512 + n*2 + (k&1)]

    const int tid  = threadIdx.x;
    const int lane = tid & 31;
    const size_t row0 = (size_t)blockIdx.x * 64;

    // --- coalesced loads into LDS (x is streamed once -> non-temporal) ---
    const float* xsrc = x + row0 * DD;
    for (int i = tid; i < (64 * DD) / 4; i += 256) {
        v4f v = __builtin_nontemporal_load(&((const v4f*)xsrc)[i]);
        int f = i * 4; int r = f >> 7; int d = f & 127;
        float* dst = &xs[r * 132 + d];
        dst[0] = v.x; dst[1] = v.y; dst[2] = v.z; dst[3] = v.w;
    }
    for (int i = tid; i < LL * DD; i += 256) {
        int l = i >> 7, d = i & 127;
        sw[l * 132 + d] = subW[i];
    }
    for (int i = tid; i < LL * OO; i += 256) {
        int k = i >> 8, n = i & 255;
        awz[(k >> 1) * 512 + n * 2 + (k & 1)] = aggW[i];
    }
    __syncthreads();

    // --- stage 1: each thread computes 4 sub-layer dots for one row ---
    {
        const int r  = tid >> 2;
        const int lb = (tid & 3) * 4;        // l base; l%4 == j -> act cycle
        const float* xr = &xs[r * 132];
        const float* w0 = &sw[(lb + 0) * 132];
        const float* w1 = &sw[(lb + 1) * 132];
        const float* w2 = &sw[(lb + 2) * 132];
        const float* w3 = &sw[(lb + 3) * 132];
        float a0 = 0.f, a1 = 0.f, a2 = 0.f, a3 = 0.f;
        #pragma unroll 4
        for (int d = 0; d < DD; ++d) {
            float xv = xr[d];
            a0 += xv * w0[d]; a1 += xv * w1[d];
            a2 += xv * w2[d]; a3 += xv * w3[d];
        }
        float z0 = a0 + subB[lb + 0];
        float z1 = a1 + subB[lb + 1];
        float z2 = a2 + subB[lb + 2];
        float z3 = a3 + subB[lb + 3];
        acts[r * 20 + lb + 0] = tanhf(z0);
        acts[r * 20 + lb + 1] = fmaxf(z1, 0.f);
        acts[r * 20 + lb + 2] = 1.0f / (1.0f + __expf(-z2));
        acts[r * 20 + lb + 3] = z3;
    }
    __syncthreads();

    // --- stage 2: act(64x16) @ agg_W(16x256) via V_WMMA_F32_16X16X4_F32 ---
    const int w    = tid >> 5;   // wave id 0..7
    const int rt   = w & 3;      // row-tile 0..3
    const int ngrp = w >> 2;     // n half 0..1
    const int h    = lane >> 4;
    const int ln   = lane & 15;
    for (int i = 0; i < 8; ++i) {
        const int n0 = (ngrp * 8 + i) * 16;
        v8f c = {};
        #pragma unroll
        for (int k0 = 0; k0 < 16; k0 += 4) {
            v2f a = *(const v2f*)&acts[(rt * 16 + ln) * 20 + k0 + 2 * h];
            v2f b = *(const v2f*)&awz[((k0 + 2 * h) >> 1) * 512 + (n0 + ln) * 2];
            c = __builtin_amdgcn_wmma_f32_16x16x4_f32(
                    false, a, false, b, (short)0, c, false, false);
        }
        #pragma unroll
        for (int r = 0; r < 8; ++r) {
            size_t gr = row0 + (size_t)(rt * 16 + r + 8 * h);
            out[gr * OO + n0 + ln] = c[r];
        }
    }
}

// ---------------------------------------------------------------------------
// Kernel 2: recurrence, in place over d_out (which holds agg).
//   state[b,:] = agg[b,t,:] + state[b,:] @ R, ys[b,t,:] = state
// 16 workgroups x 16 batch rows. R (256KB fp32) resident in LDS, pair-
// swizzled; ping-pong 16x260 state tiles. 16 waves, one 16-col strip each:
// 64 v_wmma_f32_16x16x4_f32 per wave per step.
// ---------------------------------------------------------------------------
__global__ __launch_bounds__(512) void srkan_rec_kernel(
    const float* __restrict__ Rg, float* __restrict__ out)
{
    extern __shared__ float smem[];
    float* Rl = smem;            // 65536 floats, pair-swizzled
    float* st = smem + 65536;    // 2 * 16 * 260 = 8320 floats

    const int tid  = threadIdx.x;
    const int lane = tid & 31;
    const int w    = tid >> 5;           // 0..15
    const int b0   = blockIdx.x * 16;
    const int n0   = w * 16;
    const int h    = lane >> 4;
    const int ln   = lane & 15;
    const int nn   = n0 + ln;

    // Load + swizzle R: Rl[(k>>1)*512 + n*2 + (k&1)] = R[k][n]
    for (int i = tid; i < OO * OO; i += 512) {
        int k = i >> 8, n = i & 255;
        Rl[(k >> 1) * 512 + n * 2 + (k & 1)] = Rg[i];
    }
    // state := 0 (both buffers)
    for (int i = tid; i < 8320; i += 512) st[i] = 0.f;
    __syncthreads();

    float* cur = st;
    float* nxt = st + 4160;

    for (int t = 0; t < TT; ++t) {
        // Prefetch agg (touch-once stream -> non-temporal; issued before the
        // WMMA chain so the 64-WMMA K-loop hides the global latency)
        float aggv[8]; size_t ga[8];
        #pragma unroll
        for (int r = 0; r < 8; ++r) {
            int M = r + 8 * h;
            ga[r] = (((size_t)(b0 + M)) * TT + (size_t)t) * OO + nn;
            aggv[r] = __builtin_nontemporal_load(&out[ga[r]]);
        }

        // 16x16 tile of state(16x256) @ R(256x256), K-loop of 64 f32 WMMAs
        v8f c = {};
        #pragma unroll 8
        for (int k0 = 0; k0 < 256; k0 += 4) {
            v2f a = *(const v2f*)&cur[ln * 260 + k0 + 2 * h];
            v2f b = *(const v2f*)&Rl[((k0 + 2 * h) >> 1) * 512 + nn * 2];
            c = __builtin_amdgcn_wmma_f32_16x16x4_f32(
                    false, a, false, b, (short)0, c, false, false);
        }

        // new_state = agg + state@R ; write ys in place and stage next state
        #pragma unroll
        for (int r = 0; r < 8; ++r) {
            float v = c[r] + aggv[r];
            __builtin_nontemporal_store(v, &out[ga[r]]);
            nxt[(r + 8 * h) * 260 + nn] = v;
        }
        __syncthreads();
        float* tmp = cur; cur = nxt; nxt = tmp;
    }
}

// ---------------------------------------------------------------------------
extern "C" void kernel_launch(void* const* d_in, const int* in_sizes, int n_in,
                              void* d_out, int out_size, void* d_ws, size_t ws_size,
                              hipStream_t stream) {
    const float* x    = (const float*)d_in[0];
    const float* subW = (const float*)d_in[1];
    const float* subB = (const float*)d_in[2];
    const float* aggW = (const float*)d_in[3];
    const float* R    = (const float*)d_in[4];
    float* out = (float*)d_out;

    // Phase 1: agg -> d_out (memory-bound, fully regenerated every call)
    srkan_agg_kernel<<<(BB * TT) / 64, 256, 0, stream>>>(x, subW, subB, aggW, out);

    // Phase 2: recurrence in place over d_out. Dynamic LDS:
    // 256KB swizzled R + 2 ping-pong state tiles = 295424 B (< 320KB/WGP)
    srkan_rec_kernel<<<16, 512, (65536 + 8320) * sizeof(float), stream>>>(R, out);
}